// Decoder_Layer_10230612099666
// MI455X (gfx1250) — compile-verified
//
#include <hip/hip_runtime.h>
#include <hip/hip_bf16.h>
#include <math.h>

// Problem constants (match reference)
#define M_ROWS   972
#define N_COLS   1944
#define BATCH    32
#define ROW_W    6
#define NUM_ITER 10
#define BIGV     1.0e30f
#define CAP      24                       // max column degree (Poisson(3) tail << 1e-12 at 24)
#define E_EDGES  (M_ROWS * ROW_W)         // 5832
#define BN       (BATCH * N_COLS)         // 62208
#define STACK_ELEMS ((NUM_ITER + 1) * BN) // 684288
#define LOSS_ELEMS  (NUM_ITER * BN)       // 622080
#define NB_LOSS  40                       // blocks in WMMA loss-reduction kernel

typedef __attribute__((ext_vector_type(2))) float v2f;
typedef __attribute__((ext_vector_type(8))) float v8f;

// ---------------------------------------------------------------------------
// Setup kernel 1: extract the 6 column indices of each check row (row-major
// scan => deterministic order, matches "columns ascending" slot convention).
// ---------------------------------------------------------------------------
__global__ void build_rows(const int* __restrict__ H, int* __restrict__ cols) {
  int m = blockIdx.x * blockDim.x + threadIdx.x;
  if (m >= M_ROWS) return;
  const int* row = H + (long)m * N_COLS;
  int j = 0;
  for (int n = 0; n < N_COLS; ++n) {
    if (row[n] != 0) { if (j < ROW_W) cols[m * ROW_W + j] = n; ++j; }
  }
  for (; j < ROW_W; ++j) cols[m * ROW_W + j] = 0;  // unreachable by construction
}

// ---------------------------------------------------------------------------
// Setup kernel 2: per-column edge lists (CSC) with edge id = m*6+slot,
// built scanning rows ascending => deterministic summation order.
// ---------------------------------------------------------------------------
__global__ void build_cols(const int* __restrict__ H, const int* __restrict__ cols,
                           int* __restrict__ col_deg, int* __restrict__ col_edges) {
  int n = blockIdx.x * blockDim.x + threadIdx.x;
  if (n >= N_COLS) return;
  int deg = 0;
  for (int m = 0; m < M_ROWS; ++m) {
    if (H[(long)m * N_COLS + n] != 0) {
      int base = m * ROW_W, slot = 0;
      #pragma unroll
      for (int j = 0; j < ROW_W; ++j) if (cols[base + j] == n) { slot = j; }
      if (deg < CAP) col_edges[n * CAP + deg] = base + slot;
      ++deg;
    }
  }
  col_deg[n] = (deg < CAP) ? deg : CAP;
}

// ---------------------------------------------------------------------------
// Main decoder: one workgroup per batch element, everything LDS-resident.
// Phase A (check update) and Phase B (variable marginalize) per iteration,
// separated by workgroup barriers. Writes the full [11,B,N] stack.
// Total hot-loop traffic is LDS-only; HBM sees ~10 MB total => latency bound,
// so minimizing launches + keeping state in LDS is the structural optimum.
// ---------------------------------------------------------------------------
__global__ __launch_bounds__(256) void ldpc_decode(
    const float* __restrict__ soft, const float* __restrict__ wnorm,
    const int* __restrict__ cols, const int* __restrict__ col_deg,
    const int* __restrict__ col_edges, float* __restrict__ stack) {
  __shared__ float s_soft[N_COLS];
  __shared__ float s_temp[N_COLS];
  __shared__ float s_cv[E_EDGES];
  __shared__ int   s_cols[E_EDGES];

  const int b = blockIdx.x;
  const int tid = threadIdx.x;

  // norm = softplus(w) = max(w,0) + log1p(exp(-|w|))  (matches jax.nn.softplus)
  const float w = wnorm[0];
  const float norm = fmaxf(w, 0.0f) + log1pf(expf(-fabsf(w)));

  for (int n = tid; n < N_COLS; n += 256) {
    float v = soft[b * N_COLS + n];
    s_soft[n] = v;
    s_temp[n] = v;                          // iter-0 extrinsic sum (cv == 0)
    stack[(long)b * N_COLS + n] = v;        // stack[0] = soft_input
  }
  for (int e = tid; e < E_EDGES; e += 256) { s_cv[e] = 0.0f; s_cols[e] = cols[e]; }
  __syncthreads();

  for (int t = 0; t < NUM_ITER; ++t) {
    // ---- check-node (min-sum) update: each thread owns whole rows ----
    for (int m = tid; m < M_ROWS; m += 256) {
      const int base = m * ROW_W;
      float av[ROW_W], sg[ROW_W];
      float prod = 1.0f, m1 = 3.4e38f, m2 = 3.4e38f;
      #pragma unroll
      for (int j = 0; j < ROW_W; ++j) {
        const int c = s_cols[base + j];
        const float v = s_temp[c] - s_cv[base + j];         // vc on this edge
        const float s = (v > 0.0f) ? 1.0f : ((v < 0.0f) ? -1.0f : 0.0f);
        prod *= s;
        sg[j] = s;
        const float a = fminf(fabsf(v), BIGV);              // clip(|vc|,0,BIG)
        av[j] = a;
        if (a < m1) { m2 = m1; m1 = a; } else if (a < m2) { m2 = a; }
      }
      #pragma unroll
      for (int j = 0; j < ROW_W; ++j) {
        const float r = (av[j] > m1) ? m1 : m2;             // min or 2nd-min
        s_cv[base + j] = norm * r * (prod * sg[j]);         // normalized min-sum
      }
    }
    __syncthreads();

    // ---- variable-node marginalize + emit stack[t+1] ----
    float* out = stack + (long)(t + 1) * BN + (long)b * N_COLS;
    for (int n = tid; n < N_COLS; n += 256) {
      float s = s_soft[n];
      const int deg = col_deg[n];
      const int baseE = n * CAP;
      for (int k = 0; k < deg; ++k) s += s_cv[col_edges[baseE + k]];
      s_temp[n] = s;                        // extrinsic sum for next iteration
      out[n] = s;
    }
    __syncthreads();
  }
}

// ---------------------------------------------------------------------------
// Loss reduction with V_WMMA_F32_16X16X4_F32: each wave feeds 64 CE values as
// a 16x4 f32 A-tile against an all-ones 4x16 B, accumulating D += A*1 (+C).
// The WMMA accumulator IS the reduction tree. The wave index goes through
// readfirstlane so the tile loop is SGPR-controlled: EXEC stays all-ones at
// every WMMA (ISA requirement), and loop control is pure SALU.
// ---------------------------------------------------------------------------
__global__ __launch_bounds__(256) void loss_partial(
    const float* __restrict__ stack, const int* __restrict__ labels,
    float* __restrict__ partials) {
  __shared__ float red[256];
  const float* xs = stack + BN;                       // decoded outputs, iters 1..10
  const int lane  = threadIdx.x & 31;
  // Provably wave-uniform wave index -> scalar loop, EXEC untouched.
  const int gwave = __builtin_amdgcn_readfirstlane(blockIdx.x * 8 + (threadIdx.x >> 5));
  const int nwaves = NB_LOSS * 8;
  const int ntiles = LOSS_ELEMS / 64;                 // 9720, exact

  v8f acc = {};                                       // C starts at 0
  v2f ones; ones[0] = 1.0f; ones[1] = 1.0f;

  for (int tile = gwave; tile < ntiles; tile += nwaves) {
    const int e0 = tile * 64 + lane;
    const int e1 = e0 + 32;
    // Prefetch the next tile this wave will touch (global_prefetch_b8).
    const int enext = e0 + nwaves * 64;
    if (enext < LOSS_ELEMS) __builtin_prefetch(&xs[enext], 0, 0);
    const float z0 = -xs[e0];
    const float z1 = -xs[e1];
    const float y0 = (float)labels[e0 % BN];
    const float y1 = (float)labels[e1 % BN];
    // sigmoid CE with logits z: max(z,0) - z*y + log1p(exp(-|z|))
    v2f a;
    a[0] = fmaxf(z0, 0.0f) - z0 * y0 + log1pf(expf(-fabsf(z0)));
    a[1] = fmaxf(z1, 0.0f) - z1 * y1 + log1pf(expf(-fabsf(z1)));
    acc = __builtin_amdgcn_wmma_f32_16x16x4_f32(
        /*neg_a=*/false, a, /*neg_b=*/false, ones,
        /*c_mod=*/(short)0, acc, /*reuse_a=*/false, /*reuse_b=*/false);
  }

  // D[i][j] = (row-i sum of A, accumulated), identical for all j.
  // Per-thread sum of its 8 C/D VGPRs covers M=0..7 (lanes 0-15) or M=8..15.
  float s = acc[0] + acc[1] + acc[2] + acc[3] + acc[4] + acc[5] + acc[6] + acc[7];
  red[threadIdx.x] = s;
  __syncthreads();
  if (threadIdx.x == 0) {
    float tot = 0.0f;
    for (int wv = 0; wv < 8; ++wv) tot += red[wv * 32 + 0] + red[wv * 32 + 16];
    partials[blockIdx.x] = tot;
  }
}

__global__ void final_sum(const float* __restrict__ partials, float* __restrict__ out_loss) {
  if (threadIdx.x == 0 && blockIdx.x == 0) {
    float s = 0.0f;
    for (int i = 0; i < NB_LOSS; ++i) s += partials[i];
    *out_loss = s;
  }
}

// ---------------------------------------------------------------------------
extern "C" void kernel_launch(void* const* d_in, const int* in_sizes, int n_in,
                              void* d_out, int out_size, void* d_ws, size_t ws_size,
                              hipStream_t stream) {
  const float* soft   = (const float*)d_in[0];  // [B,N] f32
  const int*   labels = (const int*)d_in[1];    // [B,N] i32
  const int*   H      = (const int*)d_in[2];    // [M,N] i32
  const float* wnorm  = (const float*)d_in[3];  // [1]   f32

  float* out = (float*)d_out;  // [stack 684288 f32][labels 62208 i32][loss 1 f32]

  char* ws = (char*)d_ws;
  int*   cols      = (int*)ws;                                   // 5832 ints
  int*   col_deg   = (int*)(ws + (size_t)E_EDGES * 4);           // 1944 ints
  int*   col_edges = (int*)(ws + (size_t)(E_EDGES + N_COLS) * 4);// 1944*CAP ints
  float* partials  = (float*)(ws + (size_t)(E_EDGES + N_COLS + N_COLS * CAP) * 4);

  build_rows<<<(M_ROWS + 255) / 256, 256, 0, stream>>>(H, cols);
  build_cols<<<(N_COLS + 255) / 256, 256, 0, stream>>>(H, cols, col_deg, col_edges);
  ldpc_decode<<<BATCH, 256, 0, stream>>>(soft, wnorm, cols, col_deg, col_edges, out);

  // labels pass-through (raw 4-byte copy into the i32 output region)
  hipMemcpyAsync((void*)(out + STACK_ELEMS), (const void*)labels,
                 (size_t)BN * sizeof(int), hipMemcpyDeviceToDevice, stream);

  loss_partial<<<NB_LOSS, 256, 0, stream>>>(out, labels, partials);
  final_sum<<<1, 1, 0, stream>>>(partials, out + STACK_ELEMS + BN);
}